// GATBlock_10153302688088
// MI455X (gfx1250) — compile-verified
//
#include <hip/hip_runtime.h>

// GATv2 forward for MI455X (gfx1250, wave32).
// N=50000 nodes, E=800000 edges, IN_DIM=256, HEADS=4, OUT_DIM=32 (FDIM=128).

#define NNODES  50000
#define NEDGES  800000
#define IN_DIMC 256
#define HEADS   4
#define ODIM    32
#define FDIM    128                 // HEADS*ODIM
#define ETOT    (NEDGES + NNODES)   // with self loops
#define NEG_SLOPE 0.2f

typedef float v2f __attribute__((ext_vector_type(2)));
typedef float v8f __attribute__((ext_vector_type(8)));

// Order-preserving float<->uint mapping so atomicMax(u32) == float max.
__device__ __forceinline__ unsigned f2ord(float f) {
  unsigned u = __float_as_uint(f);
  return (u >> 31) ? ~u : (u | 0x80000000u);
}
__device__ __forceinline__ float ord2f(unsigned o) {
  unsigned u = (o >> 31) ? (o & 0x7fffffffu) : ~o;
  return __uint_as_float(u);
}

// ---------------------------------------------------------------- init
__global__ void gat_init(float* __restrict__ out,
                         unsigned* __restrict__ m_ord,
                         float* __restrict__ denom) {
  int gid = blockIdx.x * blockDim.x + threadIdx.x;
  if (gid < NNODES * FDIM) out[gid] = 0.0f;
  if (gid < NNODES * HEADS) { m_ord[gid] = 0u; denom[gid] = 0.0f; }
}

// ------------------------------------------------- fused GEMM (WMMA fp32)
// xl = x @ W_l, xr = x @ W_r. One wave = one 16x16 output tile.
// wave id -> mTile (0..3124) x tTile (0..15); t<8 -> W_l, t>=8 -> W_r.
// A frag (16x4 f32): lanes 0-15 rows M=0..15, VGPR{0,1}=K{0,1}; lanes 16-31 K{2,3}.
// B frag (4x16 f32): mirrored (lane=N column, half selects K pair).
// C/D (16x16 f32): 8 VGPRs, row M = vgpr + 8*(lane>=16), col N = lane&15.
__global__ void gat_gemm_wmma(const float* __restrict__ x,
                              const float* __restrict__ Wl,
                              const float* __restrict__ Wr,
                              float* __restrict__ xl,
                              float* __restrict__ xr) {
  const int wave  = (blockIdx.x * blockDim.x + threadIdx.x) >> 5;
  const int lane  = threadIdx.x & 31;
  const int mTile = wave >> 4;          // 0..3124 (grid sized exactly)
  const int t     = wave & 15;          // 0..15
  const float* __restrict__ W   = (t < 8) ? Wl : Wr;
  float* __restrict__       dst = (t < 8) ? xl : xr;
  const int n0   = (t & 7) * 16;
  const int m0   = mTile * 16;
  const int half = lane >> 4;           // 0 or 1 (selects K pair / M half)
  const int l15  = lane & 15;

  const float* __restrict__ arow = x + (size_t)(m0 + l15) * IN_DIMC + 2 * half;
  const float* __restrict__ bcol = W + (size_t)(2 * half) * FDIM + n0 + l15;

  v8f c = {};
#pragma unroll 8
  for (int k = 0; k < IN_DIMC; k += 4) {
    v2f a, b;
    a.x = arow[k];
    a.y = arow[k + 1];
    b.x = bcol[(size_t)k * FDIM];
    b.y = bcol[(size_t)(k + 1) * FDIM];
    // D = A*B + C, exact fp32 (matches reference numerics)
    c = __builtin_amdgcn_wmma_f32_16x16x4_f32(false, a, false, b,
                                              (short)0, c, false, false);
  }

  float* __restrict__ orow = dst + (size_t)(m0 + half * 8) * FDIM + n0 + l15;
#pragma unroll
  for (int r = 0; r < 8; ++r) orow[(size_t)r * FDIM] = c[r];
}

// ---------------------------------------- per-edge logits + segment max
// One wave per edge; lane = channel within head. Butterfly reduce per head.
__global__ void gat_edge_logits(const int* __restrict__ ei,
                                const float* __restrict__ xl,
                                const float* __restrict__ xr,
                                const float* __restrict__ att,
                                float* __restrict__ logits,
                                unsigned* __restrict__ m_ord) {
  const int e = (blockIdx.x * blockDim.x + threadIdx.x) >> 5;
  if (e >= ETOT) return;                 // wave-uniform guard (EXEC all-ones)
  const int lane = threadIdx.x & 31;
  int s, d;
  if (e < NEDGES) { s = ei[e]; d = ei[NEDGES + e]; }
  else            { s = e - NEDGES; d = s; }       // self loop

  const float* __restrict__ xls = xl + (size_t)s * FDIM;
  const float* __restrict__ xrd = xr + (size_t)d * FDIM;

  float lg[HEADS];
#pragma unroll
  for (int h = 0; h < HEADS; ++h) {
    const int cidx = h * ODIM + lane;
    float v = xls[cidx] + xrd[cidx];
    v = (v > 0.0f) ? v : NEG_SLOPE * v;          // leaky_relu
    float p = v * att[cidx];
#pragma unroll
    for (int off = 16; off > 0; off >>= 1)        // wave32 butterfly sum
      p += __shfl_xor(p, off, 32);
    lg[h] = p;
  }
  if (lane == 0) {
#pragma unroll
    for (int h = 0; h < HEADS; ++h) {
      logits[(size_t)e * HEADS + h] = lg[h];
      atomicMax(&m_ord[(size_t)d * HEADS + h], f2ord(lg[h]));
    }
  }
}

// ----------------------------------------------- softmax denominator
__global__ void gat_edge_denom(const int* __restrict__ ei,
                               const float* __restrict__ logits,
                               const unsigned* __restrict__ m_ord,
                               float* __restrict__ denom) {
  const int gid = blockIdx.x * blockDim.x + threadIdx.x;
  if (gid >= ETOT * HEADS) return;
  const int e = gid >> 2;                 // HEADS == 4
  const int h = gid & 3;
  const int d = (e < NEDGES) ? ei[NEDGES + e] : (e - NEDGES);
  const float m  = ord2f(m_ord[(size_t)d * HEADS + h]);
  const float ex = __expf(logits[gid] - m);
  unsafeAtomicAdd(&denom[(size_t)d * HEADS + h], ex);   // global_atomic_add_f32
}

// ------------------------------------------- weighted scatter-add
__global__ void gat_edge_scatter(const int* __restrict__ ei,
                                 const float* __restrict__ xl,
                                 const float* __restrict__ logits,
                                 const unsigned* __restrict__ m_ord,
                                 const float* __restrict__ denom,
                                 float* __restrict__ out) {
  const int e = (blockIdx.x * blockDim.x + threadIdx.x) >> 5;
  if (e >= ETOT) return;
  const int lane = threadIdx.x & 31;
  int s, d;
  if (e < NEDGES) { s = ei[e]; d = ei[NEDGES + e]; }
  else            { s = e - NEDGES; d = s; }

  const float* __restrict__ xls = xl + (size_t)s * FDIM;
  float* __restrict__ od = out + (size_t)d * FDIM;
#pragma unroll
  for (int h = 0; h < HEADS; ++h) {
    const float m     = ord2f(m_ord[(size_t)d * HEADS + h]);
    const float dn    = denom[(size_t)d * HEADS + h];
    const float alpha = __expf(logits[(size_t)e * HEADS + h] - m) / (dn + 1e-16f);
    const int cidx = h * ODIM + lane;
    unsafeAtomicAdd(&od[cidx], alpha * xls[cidx]);
  }
}

// ----------------------------------------------------- bias + relu
__global__ void gat_finalize(float* __restrict__ out,
                             const float* __restrict__ bias) {
  const int gid = blockIdx.x * blockDim.x + threadIdx.x;
  if (gid >= NNODES * FDIM) return;
  const float v = out[gid] + bias[gid & (FDIM - 1)];
  out[gid] = fmaxf(v, 0.0f);
}

// ------------------------------------------------------------- launch
extern "C" void kernel_launch(void* const* d_in, const int* in_sizes, int n_in,
                              void* d_out, int out_size, void* d_ws, size_t ws_size,
                              hipStream_t stream) {
  const float* x    = (const float*)d_in[0];
  const int*   ei   = (const int*)d_in[1];
  const float* Wl   = (const float*)d_in[2];
  const float* Wr   = (const float*)d_in[3];
  const float* att  = (const float*)d_in[4];
  const float* bias = (const float*)d_in[5];
  float* out = (float*)d_out;

  // workspace layout (floats): xl | xr | logits | m_ord | denom  (~66 MB)
  float*    xl     = (float*)d_ws;
  float*    xr     = xl + (size_t)NNODES * FDIM;
  float*    logits = xr + (size_t)NNODES * FDIM;
  unsigned* m_ord  = (unsigned*)(logits + (size_t)ETOT * HEADS);
  float*    denom  = (float*)(m_ord + (size_t)NNODES * HEADS);

  {
    const int n = NNODES * FDIM;
    gat_init<<<(n + 255) / 256, 256, 0, stream>>>(out, m_ord, denom);
  }
  {
    const int waves = (NNODES / 16) * 16;          // 50,000 tiles, exact
    gat_gemm_wmma<<<(waves * 32) / 256, 256, 0, stream>>>(x, Wl, Wr, xl, xr);
  }
  {
    const long long thr = (long long)ETOT * 32;    // wave per edge
    gat_edge_logits<<<(unsigned)((thr + 255) / 256), 256, 0, stream>>>(
        ei, xl, xr, att, logits, m_ord);
  }
  {
    const int thr = ETOT * HEADS;
    gat_edge_denom<<<(thr + 255) / 256, 256, 0, stream>>>(ei, logits, m_ord, denom);
  }
  {
    const long long thr = (long long)ETOT * 32;
    gat_edge_scatter<<<(unsigned)((thr + 255) / 256), 256, 0, stream>>>(
        ei, xl, logits, m_ord, denom, out);
  }
  {
    const int n = NNODES * FDIM;
    gat_finalize<<<(n + 255) / 256, 256, 0, stream>>>(out, bias);
  }
}